// VisualGraphEmbeddingAsymetric_53171695124939
// MI455X (gfx1250) — compile-verified
//
#include <hip/hip_runtime.h>
#include <hip/hip_bf16.h>
#include <math.h>

typedef float v2f __attribute__((ext_vector_type(2)));
typedef float v8f __attribute__((ext_vector_type(8)));

#define EMB   128
#define IMG   4096
#define CAT   (IMG + EMB)
#define LDW   130            // padded LDS row stride (words) -> conflict-free b64 reads

static __device__ __forceinline__ v8f wmma4(v2f a, v2f b, v8f c) {
  return __builtin_amdgcn_wmma_f32_16x16x4_f32(false, a, false, b, (short)0, c,
                                               false, false);
}

static __device__ __forceinline__ void store_tile(const v8f& acc, float* __restrict__ C,
                                                  int m0, int hl, int n, int N, float bb) {
#pragma unroll
  for (int r = 0; r < 8; ++r)
    C[(size_t)(m0 + r + 8 * hl) * N + n] = acc[r] + bb;   // C/D: VGPR r, half hl -> row r+8*hl
}

// ---------- generic WMMA f32 GEMM:  C[M,N] = A[M,K] @ B[N,K]^T (+bias) ----------
// One wave computes a 16x64 tile; register double-buffered k-loop, chunked with
// global_prefetch_b8 for the next 256B of each stream. M%16==0, N%64==0, K%64==0.
__global__ void k_wmma_gemm_nt(const float* __restrict__ A,
                               const float* __restrict__ B,
                               const float* __restrict__ bias,
                               float* __restrict__ C,
                               int M, int N, int K) {
  const int lane = threadIdx.x & 31;
  const int wave = blockIdx.x * (blockDim.x >> 5) + (threadIdx.x >> 5);
  const int tilesN = N >> 6;
  const int tilesM = M >> 4;
  if (wave >= tilesM * tilesN) return;            // wave-uniform
  const int tm = wave / tilesN;
  const int tn = wave - tm * tilesN;
  const int lid = lane & 15;
  const int hl  = lane >> 4;
  const int m0 = tm << 4;
  const int n0 = tn << 6;

  const float* __restrict__ arow = A + (size_t)(m0 + lid) * K + 2 * hl;
  const float* __restrict__ bp0 = B + (size_t)(n0 +  0 + lid) * K + 2 * hl;
  const float* __restrict__ bp1 = B + (size_t)(n0 + 16 + lid) * K + 2 * hl;
  const float* __restrict__ bp2 = B + (size_t)(n0 + 32 + lid) * K + 2 * hl;
  const float* __restrict__ bp3 = B + (size_t)(n0 + 48 + lid) * K + 2 * hl;

  v8f acc0 = {}, acc1 = {}, acc2 = {}, acc3 = {};

  for (int kc = 0; kc < K; kc += 64) {
    if (kc + 64 < K) {                            // prefetch next chunk of all 5 streams
      __builtin_prefetch(arow + kc + 64, 0, 1);
      __builtin_prefetch(bp0  + kc + 64, 0, 1);
      __builtin_prefetch(bp1  + kc + 64, 0, 1);
      __builtin_prefetch(bp2  + kc + 64, 0, 1);
      __builtin_prefetch(bp3  + kc + 64, 0, 1);
    }
    // software pipeline within the chunk: load step s while issuing WMMAs of s-1
    v2f a_n  = *(const v2f*)(arow + kc);
    v2f b0_n = *(const v2f*)(bp0 + kc);
    v2f b1_n = *(const v2f*)(bp1 + kc);
    v2f b2_n = *(const v2f*)(bp2 + kc);
    v2f b3_n = *(const v2f*)(bp3 + kc);
    const int klim = kc + 64;
    for (int k0 = kc + 4; k0 < klim; k0 += 4) {
      v2f a_c = a_n, b0_c = b0_n, b1_c = b1_n, b2_c = b2_n, b3_c = b3_n;
      a_n  = *(const v2f*)(arow + k0);
      b0_n = *(const v2f*)(bp0 + k0);
      b1_n = *(const v2f*)(bp1 + k0);
      b2_n = *(const v2f*)(bp2 + k0);
      b3_n = *(const v2f*)(bp3 + k0);
      acc0 = wmma4(a_c, b0_c, acc0);
      acc1 = wmma4(a_c, b1_c, acc1);
      acc2 = wmma4(a_c, b2_c, acc2);
      acc3 = wmma4(a_c, b3_c, acc3);
    }
    acc0 = wmma4(a_n, b0_n, acc0);                // chunk epilogue
    acc1 = wmma4(a_n, b1_n, acc1);
    acc2 = wmma4(a_n, b2_n, acc2);
    acc3 = wmma4(a_n, b3_n, acc3);
  }

  const float bb0 = bias ? bias[n0 +  0 + lid] : 0.0f;
  const float bb1 = bias ? bias[n0 + 16 + lid] : 0.0f;
  const float bb2 = bias ? bias[n0 + 32 + lid] : 0.0f;
  const float bb3 = bias ? bias[n0 + 48 + lid] : 0.0f;
  store_tile(acc0, C, m0, hl, n0 +  0 + lid, N, bb0);
  store_tile(acc1, C, m0, hl, n0 + 16 + lid, N, bb1);
  store_tile(acc2, C, m0, hl, n0 + 32 + lid, N, bb2);
  store_tile(acc3, C, m0, hl, n0 + 48 + lid, N, bb3);
}

// ---------- GCN-layer GEMM: C[M,128] = A[M,128] @ W[128,128]^T, W staged in LDS ----
// Block = 256 threads = 8 waves = 4 M-tiles x 2 N-tiles (64x128 output tile).
// W stored padded (stride 130 words) -> lanes read ds at bank-stride 2, conflict-free.
__global__ void __launch_bounds__(256)
k_wmma_gemm_gcn(const float* __restrict__ A, const float* __restrict__ W,
                float* __restrict__ C, int M) {
  __shared__ float sw[128 * LDW];
  for (int idx = threadIdx.x; idx < 128 * 32; idx += 256) {   // 16384 floats, float4 each
    const int nrow = idx >> 5;
    const int kc = (idx & 31) * 4;
    const float4 v = *(const float4*)(W + (size_t)nrow * 128 + kc);
    float* d = sw + nrow * LDW + kc;
    d[0] = v.x; d[1] = v.y; d[2] = v.z; d[3] = v.w;
  }
  __syncthreads();

  const int lane = threadIdx.x & 31;
  const int wv = threadIdx.x >> 5;                 // 0..7
  const int mt = blockIdx.x * 4 + (wv >> 1);
  if (mt * 16 >= M) return;                        // wave-uniform, after barrier
  const int n0 = (wv & 1) * 64;
  const int lid = lane & 15;
  const int hl = lane >> 4;
  const int m0 = mt << 4;

  const float* __restrict__ arow = A + (size_t)(m0 + lid) * EMB + 2 * hl;
  const float* __restrict__ s0 = sw + (n0 +  0 + lid) * LDW + 2 * hl;
  const float* __restrict__ s1 = sw + (n0 + 16 + lid) * LDW + 2 * hl;
  const float* __restrict__ s2 = sw + (n0 + 32 + lid) * LDW + 2 * hl;
  const float* __restrict__ s3 = sw + (n0 + 48 + lid) * LDW + 2 * hl;

  v8f acc0 = {}, acc1 = {}, acc2 = {}, acc3 = {};
  v2f a_n = *(const v2f*)(arow);
#pragma unroll 8
  for (int k0 = 4; k0 < EMB; k0 += 4) {
    v2f a_c = a_n;
    a_n = *(const v2f*)(arow + k0);
    v2f b0 = *(const v2f*)(s0 + k0 - 4);
    v2f b1 = *(const v2f*)(s1 + k0 - 4);
    v2f b2 = *(const v2f*)(s2 + k0 - 4);
    v2f b3 = *(const v2f*)(s3 + k0 - 4);
    acc0 = wmma4(a_c, b0, acc0);
    acc1 = wmma4(a_c, b1, acc1);
    acc2 = wmma4(a_c, b2, acc2);
    acc3 = wmma4(a_c, b3, acc3);
  }
  {
    v2f b0 = *(const v2f*)(s0 + 124);
    v2f b1 = *(const v2f*)(s1 + 124);
    v2f b2 = *(const v2f*)(s2 + 124);
    v2f b3 = *(const v2f*)(s3 + 124);
    acc0 = wmma4(a_n, b0, acc0);
    acc1 = wmma4(a_n, b1, acc1);
    acc2 = wmma4(a_n, b2, acc2);
    acc3 = wmma4(a_n, b3, acc3);
  }
  store_tile(acc0, C, m0, hl, n0 +  0 + lid, EMB, 0.0f);
  store_tile(acc1, C, m0, hl, n0 + 16 + lid, EMB, 0.0f);
  store_tile(acc2, C, m0, hl, n0 + 32 + lid, EMB, 0.0f);
  store_tile(acc3, C, m0, hl, n0 + 48 + lid, EMB, 0.0f);
}

// ---------------- small helper kernels ----------------------------------------
__global__ void k_fill(float* p, float v, int n) {
  int i = blockIdx.x * blockDim.x + threadIdx.x;
  if (i < n) p[i] = v;
}

__global__ void k_deg_acc(const int* __restrict__ col, const float* __restrict__ ew,
                          float* __restrict__ deg, int E) {
  int e = blockIdx.x * blockDim.x + threadIdx.x;
  if (e < E) unsafeAtomicAdd(&deg[col[e]], ew[e]);
}

__global__ void k_rsqrt_inplace(float* p, int n) {
  int i = blockIdx.x * blockDim.x + threadIdx.x;
  if (i < n) p[i] = rsqrtf(p[i]);
}

__global__ void k_gather(const int* __restrict__ types, const float* __restrict__ emb,
                         float* __restrict__ x, int n) {
  int t = blockIdx.x * blockDim.x + threadIdx.x;
  int i = t >> 5;
  if (i >= n) return;
  int c = (t & 31) * 4;
  *(float4*)(x + (size_t)i * EMB + c) =
      *(const float4*)(emb + (size_t)types[i] * EMB + c);
}

__global__ void k_selfinit(const float* __restrict__ dis, const float* __restrict__ xw,
                           const float* __restrict__ bias, float* __restrict__ agg, int n) {
  int t = blockIdx.x * blockDim.x + threadIdx.x;
  int i = t >> 5;
  if (i >= n) return;
  int c = (t & 31) * 4;
  float d2 = dis[i]; d2 *= d2;
  float4 v = *(const float4*)(xw + (size_t)i * EMB + c);
  float4 b = *(const float4*)(bias + c);
  float4 o;
  o.x = fmaf(d2, v.x, b.x); o.y = fmaf(d2, v.y, b.y);
  o.z = fmaf(d2, v.z, b.z); o.w = fmaf(d2, v.w, b.w);
  *(float4*)(agg + (size_t)i * EMB + c) = o;
}

__global__ void k_edge_agg(const int* __restrict__ row, const int* __restrict__ col,
                           const float* __restrict__ ew, const float* __restrict__ dis,
                           const float* __restrict__ xw, float* __restrict__ agg, int E) {
  int t = blockIdx.x * blockDim.x + threadIdx.x;
  int e = t >> 5;
  if (e >= E) return;
  int c = (t & 31) * 4;
  int r = row[e], cl = col[e];
  float nrm = dis[r] * ew[e] * dis[cl];
  float4 v = *(const float4*)(xw + (size_t)r * EMB + c);
  float* dst = agg + (size_t)cl * EMB + c;
  unsafeAtomicAdd(dst + 0, nrm * v.x);
  unsafeAtomicAdd(dst + 1, nrm * v.y);
  unsafeAtomicAdd(dst + 2, nrm * v.z);
  unsafeAtomicAdd(dst + 3, nrm * v.w);
}

__global__ void k_relu(const float* __restrict__ in, float* __restrict__ out, int n) {
  int i = blockIdx.x * blockDim.x + threadIdx.x;
  if (i < n) out[i] = fmaxf(in[i], 0.0f);
}

__global__ void k_pool_acc(const int* __restrict__ batch, const float* __restrict__ xg,
                           float* __restrict__ pooled, float* __restrict__ cnt, int n) {
  int t = blockIdx.x * blockDim.x + threadIdx.x;
  int i = t >> 5;
  if (i >= n) return;
  int c = (t & 31) * 4;
  int g = batch[i];
  float4 v = *(const float4*)(xg + (size_t)i * EMB + c);
  float* dst = pooled + (size_t)g * EMB + c;
  unsafeAtomicAdd(dst + 0, v.x);
  unsafeAtomicAdd(dst + 1, v.y);
  unsafeAtomicAdd(dst + 2, v.z);
  unsafeAtomicAdd(dst + 3, v.w);
  if ((t & 31) == 0) unsafeAtomicAdd(&cnt[g], 1.0f);
}

__global__ void k_pool_div(float* pooled, const float* __restrict__ cnt, int G) {
  int t = blockIdx.x * blockDim.x + threadIdx.x;
  if (t < G * EMB) pooled[t] /= fmaxf(cnt[t >> 7], 1.0f);
}

__global__ void k_xcat(const float* __restrict__ xi, const float* __restrict__ pooled,
                       float* __restrict__ xcat, int G) {
  int t = blockIdx.x * blockDim.x + threadIdx.x;
  if (t >= G * CAT) return;
  int r = t / CAT, c = t - r * CAT;
  xcat[t] = (c < IMG) ? xi[(size_t)r * IMG + c] : pooled[(size_t)r * EMB + (c - IMG)];
}

__global__ void k_rownorm(float* x) {
  __shared__ float s[128];
  int r = blockIdx.x, t = threadIdx.x;
  float v = x[(size_t)r * EMB + t];
  s[t] = v * v;
  __syncthreads();
  for (int o = 64; o > 0; o >>= 1) {
    if (t < o) s[t] += s[t + o];
    __syncthreads();
  }
  x[(size_t)r * EMB + t] = v * rsqrtf(s[0]);
}

// ---------------- host side ---------------------------------------------------
static inline int cdiv(int a, int b) { return (a + b - 1) / b; }

static void launch_gemm(const float* A, const float* B, const float* bias, float* C,
                        int M, int N, int K, hipStream_t s) {
  int waves = (M / 16) * (N / 64);
  hipLaunchKernelGGL(k_wmma_gemm_nt, dim3(cdiv(waves, 4)), dim3(128), 0, s,
                     A, B, bias, C, M, N, K);
}

extern "C" void kernel_launch(void* const* d_in, const int* in_sizes, int n_in,
                              void* d_out, int out_size, void* d_ws, size_t ws_size,
                              hipStream_t stream) {
  const float* images    = (const float*)d_in[0];
  const int*   node_types= (const int*)  d_in[1];
  const int*   edge_index= (const int*)  d_in[2];
  const float* edge_attr = (const float*)d_in[3];
  const int*   batch     = (const int*)  d_in[4];
  const float* emb       = (const float*)d_in[5];
  const float* W_img     = (const float*)d_in[6];
  const float* b_img     = (const float*)d_in[7];
  const float* W1 = (const float*)d_in[8],  *b1 = (const float*)d_in[9];
  const float* W2 = (const float*)d_in[10], *b2 = (const float*)d_in[11];
  const float* W3 = (const float*)d_in[12], *b3 = (const float*)d_in[13];
  const float* Wi = (const float*)d_in[14], *bi = (const float*)d_in[15];
  const float* Wc = (const float*)d_in[16], *bc = (const float*)d_in[17];

  const int n = in_sizes[1];                 // 50000 nodes (multiple of 16)
  const int E = in_sizes[3];                 // 800000 edges
  const int G = in_sizes[0] / IMG;           // 64 graphs

  const int* erow = edge_index;
  const int* ecol = edge_index + E;

  float* ws = (float*)d_ws;
  size_t off = 0;
  float* dis   = ws + off; off += (size_t)((n + 31) & ~31);
  float* bufA  = ws + off; off += (size_t)n * EMB;
  float* bufW  = ws + off; off += (size_t)n * EMB;
  float* bufB  = ws + off; off += (size_t)n * EMB;
  float* xi    = ws + off; off += (size_t)G * IMG;
  float* xcat  = ws + off; off += (size_t)G * CAT;
  float* pooled= ws + off; off += (size_t)G * EMB;
  float* cnt   = ws + off; off += (size_t)G;

  float* oi = (float*)d_out;
  float* oc = oi + (size_t)G * EMB;

  const int TB = 256;

  // 1) dis = rsqrt(deg + 1)
  hipLaunchKernelGGL(k_fill, dim3(cdiv(n, TB)), dim3(TB), 0, stream, dis, 1.0f, n);
  hipLaunchKernelGGL(k_deg_acc, dim3(cdiv(E, TB)), dim3(TB), 0, stream, ecol, edge_attr, dis, E);
  hipLaunchKernelGGL(k_rsqrt_inplace, dim3(cdiv(n, TB)), dim3(TB), 0, stream, dis, n);

  // 2) xi = images @ W_img^T + b_img
  launch_gemm(images, W_img, b_img, xi, G, IMG, IMG, stream);

  // 3) node embeddings
  hipLaunchKernelGGL(k_gather, dim3(cdiv(n * 32, TB)), dim3(TB), 0, stream,
                     node_types, emb, bufA, n);

  // 4) three GCN layers (LDS-staged weight GEMM)
  const float* Ws[3] = {W1, W2, W3};
  const float* bs[3] = {b1, b2, b3};
  for (int l = 0; l < 3; ++l) {
    hipLaunchKernelGGL(k_wmma_gemm_gcn, dim3(cdiv(n / 16, 4)), dim3(256), 0, stream,
                       bufA, Ws[l], bufW, n);
    hipLaunchKernelGGL(k_selfinit, dim3(cdiv(n * 32, TB)), dim3(TB), 0, stream,
                       dis, bufW, bs[l], bufB, n);
    hipLaunchKernelGGL(k_edge_agg, dim3(cdiv(E * 32, TB)), dim3(TB), 0, stream,
                       erow, ecol, edge_attr, dis, bufW, bufB, E);
    if (l < 2)
      hipLaunchKernelGGL(k_relu, dim3(cdiv(n * EMB, TB)), dim3(TB), 0, stream,
                         bufB, bufA, n * EMB);
  }

  // 5) mean pooling
  hipLaunchKernelGGL(k_fill, dim3(cdiv(G * EMB, TB)), dim3(TB), 0, stream, pooled, 0.0f, G * EMB);
  hipLaunchKernelGGL(k_fill, dim3(1), dim3(TB), 0, stream, cnt, 0.0f, G);
  hipLaunchKernelGGL(k_pool_acc, dim3(cdiv(n * 32, TB)), dim3(TB), 0, stream,
                     batch, bufB, pooled, cnt, n);
  hipLaunchKernelGGL(k_pool_div, dim3(cdiv(G * EMB, TB)), dim3(TB), 0, stream, pooled, cnt, G);

  // 6) oi
  launch_gemm(xi, Wi, bi, oi, G, EMB, IMG, stream);
  hipLaunchKernelGGL(k_rownorm, dim3(G), dim3(128), 0, stream, oi);

  // 7) oc
  hipLaunchKernelGGL(k_xcat, dim3(cdiv(G * CAT, TB)), dim3(TB), 0, stream, xi, pooled, xcat, G);
  launch_gemm(xcat, Wc, bc, oc, G, EMB, CAT, stream);
  hipLaunchKernelGGL(k_rownorm, dim3(G), dim3(128), 0, stream, oc);
}